// TransducerLoss_46325517254827
// MI455X (gfx1250) — compile-verified
//
#include <hip/hip_runtime.h>
#include <math.h>

// Problem dimensions (fixed by the reference's setup_inputs()).
namespace {
constexpr int Bn  = 16;
constexpr int Tn  = 200;
constexpr int Un  = 100;
constexpr int U1n = 101;   // U + 1
constexpr int V1n = 129;   // vocab + blank
constexpr int ROWS = Bn * Tn * U1n;          // 323,200 softmax rows
constexpr int BLANK_ELEMS = Tn * U1n;        // 20,200 floats / batch
constexpr int LABEL_ELEMS = Tn * Un;         // 20,000 floats / batch
}

// ---------------------------------------------------------------------------
// Wave32 reductions
// ---------------------------------------------------------------------------
__device__ __forceinline__ float wave_max(float v) {
#pragma unroll
  for (int off = 16; off > 0; off >>= 1) v = fmaxf(v, __shfl_xor(v, off, 32));
  return v;
}
__device__ __forceinline__ float wave_sum(float v) {
#pragma unroll
  for (int off = 16; off > 0; off >>= 1) v += __shfl_xor(v, off, 32);
  return v;
}

// ---------------------------------------------------------------------------
// Phase 1: streaming log-softmax over V1=129 (memory-bound, ~167MB once).
// One wave32 per (b,t,u) row; NT loads since data is touched exactly once.
//   blank_lp[b,t,u] = logits[b,t,u,0]           - lse
//   label_lp[b,t,u] = logits[b,t,u,labels[b,u]] - lse    (u < U)
// ---------------------------------------------------------------------------
__global__ __launch_bounds__(256) void rnnt_logsoftmax(
    const float* __restrict__ logits, const int* __restrict__ labels,
    float* __restrict__ blank_lp, float* __restrict__ label_lp) {
  const int wave = blockIdx.x * (blockDim.x >> 5) + (threadIdx.x >> 5);
  const int lane = threadIdx.x & 31;
  if (wave >= ROWS) return;

  const int u  = wave % U1n;
  const int bt = wave / U1n;
  const int t  = bt % Tn;
  const int b  = bt / Tn;

  const float* base = logits + (size_t)wave * V1n;
  // Each load instruction covers a contiguous, coalesced 128B span.
  const float x0 = __builtin_nontemporal_load(base + lane);
  const float x1 = __builtin_nontemporal_load(base + lane + 32);
  const float x2 = __builtin_nontemporal_load(base + lane + 64);
  const float x3 = __builtin_nontemporal_load(base + lane + 96);
  const float ex = __builtin_nontemporal_load(base + 128);  // wave-uniform

  float m = wave_max(fmaxf(fmaxf(x0, x1), fmaxf(x2, x3)));
  m = fmaxf(m, ex);
  float s = wave_sum(__expf(x0 - m) + __expf(x1 - m) +
                     __expf(x2 - m) + __expf(x3 - m));
  s += __expf(ex - m);                    // add the 129th term once
  const float lse = m + __logf(s);

  const float blank = __shfl(x0, 0, 32) - lse;   // col 0 lives in lane 0's x0
  if (lane == 0) blank_lp[wave] = blank;

  if (u < Un) {
    const int lab = labels[b * Un + u];   // in [1, 129), wave-uniform
    float lv;
    if (lab == 128) {
      lv = ex;
    } else {
      const int slot = lab >> 5;          // which of x0..x3
      const int src  = lab & 31;          // which lane
      const float ch = slot == 0 ? x0 : slot == 1 ? x1 : slot == 2 ? x2 : x3;
      lv = __shfl(ch, src, 32);
    }
    if (lane == 0)
      label_lp[(size_t)(b * Tn + t) * Un + u] = lv - lse;
  }
}

// ---------------------------------------------------------------------------
// Phase 2: per-batch forward lattice (anti-diagonal wavefront), one wave32
// per batch (16 WGPs busy, zero barriers). The full per-batch blank/label
// tables (160.8 KB, fits the WGP's 320 KB LDS) are staged with CDNA5
// ASYNCcnt-tracked async-to-LDS copies; the DP then runs from LDS+registers.
// ---------------------------------------------------------------------------
__device__ __forceinline__ void async_copy_b128(unsigned lds_byte_off,
                                                const float* gsrc) {
  // GLOBAL_LOAD_ASYNC_TO_LDS_B128: VDST = per-lane LDS byte address,
  // VADDR = per-lane 64-bit global address. Tracked by ASYNCcnt.
  asm volatile("global_load_async_to_lds_b128 %0, %1, off"
               :: "v"(lds_byte_off), "v"(gsrc)
               : "memory");
}
__device__ __forceinline__ void wait_async0() {
  asm volatile("s_wait_asynccnt 0x0" ::: "memory");
}

// Branchless log-add-exp. d = mn-mx <= 0, so log(1+exp(d)) in [0, ln2];
// relative to |mx| (up to ~1e2-1e3) the log1p->log(1+x) substitution is far
// below fp32 ulp of the accumulated log-prob. Guard -inf/-inf -> NaN.
__device__ __forceinline__ float laddexp(float a, float b) {
  const float mx = fmaxf(a, b);
  const float mn = fminf(a, b);
  const float r  = mx + __logf(1.0f + __expf(mn - mx));
  return (mx == -INFINITY) ? -INFINITY : r;
}

// alpha[t][u] for cell on diagonal d (t = d - u):
//   below = alpha[t-1][u] (old diag @ u), left = alpha[t][u-1] (old diag @ u-1)
__device__ __forceinline__ float dp_cell(int d, int u, float below, float left,
                                         const float* __restrict__ bS,
                                         const float* __restrict__ lS) {
  const int t = d - u;
  const bool valid = (u <= Un) && (t >= 0) && (t < Tn);
  float fb = -INFINITY, fl = -INFINITY;
  if (valid && t >= 1) fb = below + bS[(t - 1) * U1n + u];
  if (valid && u >= 1) fl = left + lS[t * Un + (u - 1)];
  const float r = laddexp(fb, fl);
  return valid ? r : -INFINITY;
}

__global__ __launch_bounds__(32) void rnnt_alpha(
    const float* __restrict__ blank_lp, const float* __restrict__ label_lp,
    const int* __restrict__ logit_lens, const int* __restrict__ y_lens,
    float* __restrict__ losses) {
  extern __shared__ float smem[];          // dynamic LDS starts at offset 0
  float* bS = smem;                        // [Tn*U1n]
  float* lS = smem + BLANK_ELEMS;          // [Tn*Un]
  const unsigned labelLdsBase = (unsigned)(BLANK_ELEMS * sizeof(float));

  const int b    = blockIdx.x;
  const int lane = threadIdx.x;

  // ---- Stage blank_lp[b] (80,800B) and label_lp[b] (80,000B) into LDS ----
  const float* gb = blank_lp + (size_t)b * BLANK_ELEMS;  // 16B-aligned
  const float* gl = label_lp + (size_t)b * LABEL_ELEMS;  // 16B-aligned
  const int nb4 = BLANK_ELEMS / 4;   // 5050 x 16B
  const int nl4 = LABEL_ELEMS / 4;   // 5000 x 16B
  for (int i = lane; i < nb4; i += 32)
    async_copy_b128((unsigned)(i * 16), gb + i * 4);
  for (int i = lane; i < nl4; i += 32)
    async_copy_b128(labelLdsBase + (unsigned)(i * 16), gl + i * 4);

  const int tl    = logit_lens[b] - 1;     // scalar loads overlap the DMA
  const int yl    = y_lens[b];
  const int dsave = tl + yl;               // diagonal holding alpha[tl][yl]

  wait_async0();

  // ---- Wavefront DP: lane owns u = lane + 32c, c = 0..3 ----
  float o0 = -INFINITY, o1 = -INFINITY, o2 = -INFINITY, o3 = -INFINITY;
  if (lane == 0) o0 = 0.0f;                // d=0: alpha[0][0] = 0

  // Only diagonals 1..dsave are needed for this batch (dsave <= 299).
  for (int d = 1; d <= dsave; ++d) {
    // left neighbor (u-1) from previous diagonal
    float l0 = __shfl_up(o0, 1, 32);
    float l1 = __shfl_up(o1, 1, 32);
    float l2 = __shfl_up(o2, 1, 32);
    float l3 = __shfl_up(o3, 1, 32);
    const float h0 = __shfl(o0, 31, 32);   // chunk-boundary fixups
    const float h1 = __shfl(o1, 31, 32);
    const float h2 = __shfl(o2, 31, 32);
    if (lane == 0) { l0 = -INFINITY; l1 = h0; l2 = h1; l3 = h2; }

    const float n0 = dp_cell(d, lane +  0, o0, l0, bS, lS);
    const float n1 = dp_cell(d, lane + 32, o1, l1, bS, lS);
    const float n2 = dp_cell(d, lane + 64, o2, l2, bS, lS);
    const float n3 = dp_cell(d, lane + 96, o3, l3, bS, lS);
    o0 = n0; o1 = n1; o2 = n2; o3 = n3;
  }

  // After the last iteration (d == dsave), alpha[tl][yl] sits at u == yl.
  if (lane == (yl & 31)) {
    const int c = yl >> 5;
    const float alpha_final = c == 0 ? o0 : c == 1 ? o1 : c == 2 ? o2 : o3;
    const float final_lp = alpha_final + bS[tl * U1n + yl];
    losses[b] = (-final_lp) / (float)yl;   // per-batch loss / y_len
  }
}

// ---------------------------------------------------------------------------
// Phase 3: mean over batch -> scalar output.
// ---------------------------------------------------------------------------
__global__ __launch_bounds__(32) void rnnt_reduce(
    const float* __restrict__ losses, float* __restrict__ out) {
  const int lane = threadIdx.x;
  float v = (lane < Bn) ? losses[lane] : 0.0f;
  v = wave_sum(v);
  if (lane == 0) out[0] = v / (float)Bn;
}

// ---------------------------------------------------------------------------
extern "C" void kernel_launch(void* const* d_in, const int* in_sizes, int n_in,
                              void* d_out, int out_size, void* d_ws,
                              size_t ws_size, hipStream_t stream) {
  const float* logits     = (const float*)d_in[0];
  const int*   labels     = (const int*)d_in[1];
  const int*   logit_lens = (const int*)d_in[2];
  const int*   y_lens     = (const int*)d_in[3];

  float* blank_lp = (float*)d_ws;                         // 323,200 floats
  float* label_lp = blank_lp + (size_t)Bn * BLANK_ELEMS;  // 320,000 floats
  float* losses   = label_lp + (size_t)Bn * LABEL_ELEMS;  // 16 floats

  const int waves_per_block = 8;                          // 256 threads
  const int blocks = (ROWS + waves_per_block - 1) / waves_per_block;
  rnnt_logsoftmax<<<blocks, 256, 0, stream>>>(logits, labels, blank_lp,
                                              label_lp);

  const size_t ldsBytes = (size_t)(BLANK_ELEMS + LABEL_ELEMS) * sizeof(float);
  rnnt_alpha<<<Bn, 32, ldsBytes, stream>>>(blank_lp, label_lp, logit_lens,
                                           y_lens, losses);

  rnnt_reduce<<<1, 32, 0, stream>>>(losses, (float*)d_out);
}